// GCN_71064528879668
// MI455X (gfx1250) — compile-verified
//
#include <hip/hip_runtime.h>
#include <math.h>

#define NNODES 50000
#define NEDGES 800000

typedef float v2f __attribute__((ext_vector_type(2)));
typedef float v8f __attribute__((ext_vector_type(8)));
typedef unsigned int v4u __attribute__((ext_vector_type(4)));
typedef int v8i __attribute__((ext_vector_type(8)));
typedef int v4i __attribute__((ext_vector_type(4)));

// ---------------- degree / norm kernels ----------------

__global__ void k_init_deg(float* deg, int n) {
    int i = blockIdx.x * blockDim.x + threadIdx.x;
    if (i < n) deg[i] = 1.0f;  // self-loop
}

__global__ void k_count_deg(float* deg, const int* __restrict__ dst, int e) {
    int i = blockIdx.x * blockDim.x + threadIdx.x;
    if (i < e) atomicAdd(&deg[dst[i]], 1.0f);
}

__global__ void k_rsqrt_deg(float* deg, int n) {
    int i = blockIdx.x * blockDim.x + threadIdx.x;
    if (i < n) deg[i] = rsqrtf(deg[i]);  // deg >= 1 always
}

// ---------------- FP32 WMMA GEMM: C[nrows x NC] = A[nrows x K] @ W[K x NC] ----------------
// Block: 256 threads = 8 waves. Each wave: one 16x64 C tile (4 WMMA accumulators).
// Weight panel W[:, n0:n0+64] staged row-major in LDS by the Tensor Data Mover (one
// tensor_load_to_lds per block; TDM ignores EXEC so a single-lane guard issues it once),
// completion via TENSORcnt. A/B fragments per CDNA5 ISA 32-bit 16x4 layout:
// lanes0-15 hold K=k0(+1), lanes16-31 hold K=k0+2(+3).

__global__ void __launch_bounds__(256)
k_gemm_wmma(const float* __restrict__ A, const float* __restrict__ W,
            const float* __restrict__ bias, float* __restrict__ C,
            int nrows, int K, int NC, int doRelu) {
    extern __shared__ float ldsW[];  // K*64 floats, row-major: ldsW[k*64 + n]
    const int tid = threadIdx.x;
    const int n0  = blockIdx.y * 64;

#if __has_builtin(__builtin_amdgcn_tensor_load_to_lds)
    if (tid == 0) {
        // ---- Tensor DMA descriptor (D#): 2D tile, 64 wide x K rows, f32 ----
        unsigned long long ga = (unsigned long long)(uintptr_t)(W + n0);  // tile start
        unsigned int lds0 = (unsigned int)(uintptr_t)ldsW;                // LDS byte offset (low 32 of flat)
        v4u g0;
        g0[0] = 1u;                                             // count=1, is_restore=0, gather=0
        g0[1] = lds0;                                           // lds_addr
        g0[2] = (unsigned int)ga;                               // global_addr[31:0]
        g0[3] = (unsigned int)((ga >> 32) & 0x01FFFFFFu)        // global_addr[56:32]
              | 0x80000000u;                                    // type=2 ("image")
        const unsigned int td0 = 64u, td1 = (unsigned int)K;    // tensor dims (tile-exact)
        const unsigned int tl0 = 64u, tl1 = (unsigned int)K;    // tile dims
        const unsigned int st0 = (unsigned int)NC;              // dim0 stride (elements)
        v8i g1;
        g1[0] = (int)(2u << 16);                                // wg_mask=0, data_size=2 (4B)
        g1[1] = (int)((td0 & 0xFFFFu) << 16);                   // tensor_dim0[15:0] @bits63:48
        g1[2] = (int)((td0 >> 16) | ((td1 & 0xFFFFu) << 16));   // tensor_dim0 hi | tensor_dim1 lo
        g1[3] = (int)((td1 >> 16) | (tl0 << 16));               // tensor_dim1 hi | tile_dim0
        g1[4] = (int)(tl1 & 0xFFFFu);                           // tile_dim1 (tile_dim2=0)
        g1[5] = (int)st0;                                       // tensor_dim0_stride[31:0]
        g1[6] = 0;                                              // stride hi | dim1_stride lo
        g1[7] = 0;
        v4i g2 = {0, 0, 0, 0};
        v4i g3 = {0, 0, 0, 0};
        v8i g4 = {0, 0, 0, 0, 0, 0, 0, 0};                      // unused slot (6-arg toolchain)
        __builtin_amdgcn_tensor_load_to_lds(g0, g1, g2, g3, g4, 0);
        __builtin_amdgcn_s_wait_tensorcnt(0);
    }
#else
    for (int i = tid; i < K * 64; i += 256) {
        int k = i >> 6, n = i & 63;
        ldsW[i] = W[k * NC + n0 + n];
    }
#endif
    __syncthreads();

    const int wave   = tid >> 5;
    const int lane   = tid & 31;
    const int lane16 = lane & 15;
    const int hi     = lane >> 4;          // 0 or 1
    const int mtile  = blockIdx.x * 8 + wave;
    if (mtile * 16 >= nrows) return;       // wave-uniform exit: EXEC stays all-ones
    const int m0 = mtile * 16;

    v8f acc[4] = {};

    const float* arow = A + (size_t)(m0 + lane16) * K;
    for (int k0 = 0; k0 < K; k0 += 4) {
        const int kk = k0 + 2 * hi;
        v2f a = *(const v2f*)(arow + kk);                   // A[m][kk], A[m][kk+1]
        const float* bb = ldsW + kk * 64 + lane16;
#pragma unroll
        for (int j = 0; j < 4; ++j) {
            v2f b;
            b.x = bb[j * 16];                               // W[kk][n]
            b.y = bb[j * 16 + 64];                          // W[kk+1][n]
            acc[j] = __builtin_amdgcn_wmma_f32_16x16x4_f32(
                false, a, false, b, (short)0, acc[j], false, false);
        }
    }

#pragma unroll
    for (int j = 0; j < 4; ++j) {
        const int col = n0 + j * 16 + lane16;
        const float bv = bias ? bias[col] : 0.0f;
#pragma unroll
        for (int v = 0; v < 8; ++v) {
            const int row = m0 + hi * 8 + v;                // C/D layout: VGPRv = M=v / M=v+8
            float val = acc[j][v] + bv;
            if (doRelu) val = fmaxf(val, 0.0f);
            C[(size_t)row * NC + col] = val;
        }
    }
}

// ---------------- aggregation kernels ----------------

// out4[idx] = h4[idx] * dinv[i]^2 ; float4 over N*F/4 elements, fdivshift = log2(F/4)
__global__ void k_agg_init(const float4* __restrict__ h4, const float* __restrict__ dinv,
                           float4* __restrict__ out4, int n4, int fdivshift) {
    int idx = blockIdx.x * blockDim.x + threadIdx.x;
    if (idx < n4) {
        float di = dinv[idx >> fdivshift];
        float s = di * di;
        float4 v = h4[idx];
        v.x *= s; v.y *= s; v.z *= s; v.w *= s;
        out4[idx] = v;
    }
}

__global__ void k_agg_edges(const float* __restrict__ h, const float* __restrict__ dinv,
                            const int* __restrict__ src, const int* __restrict__ dst,
                            float* __restrict__ out, int e, int F) {
    int wid  = (blockIdx.x * blockDim.x + threadIdx.x) >> 5;  // one wave per edge
    int lane = threadIdx.x & 31;
    if (wid >= e) return;
    int s = src[wid], d = dst[wid];
    float norm = dinv[s] * dinv[d];
    const float4* hs = (const float4*)(h + (size_t)s * F);
    float* od = out + (size_t)d * F;
    for (int q = lane; q * 4 < F; q += 32) {                  // b128 gather, scalar f32 atomics
        float4 v = hs[q];
        atomicAdd(&od[q * 4 + 0], v.x * norm);
        atomicAdd(&od[q * 4 + 1], v.y * norm);
        atomicAdd(&od[q * 4 + 2], v.z * norm);
        atomicAdd(&od[q * 4 + 3], v.w * norm);
    }
}

// x4[idx] = relu(x4[idx] + bias4[idx & bm4]) ; float4 over N*F/4
__global__ void k_bias_relu(float4* __restrict__ x4, const float4* __restrict__ bias4,
                            int n4, int bm4, int doRelu) {
    int idx = blockIdx.x * blockDim.x + threadIdx.x;
    if (idx < n4) {
        float4 v = x4[idx];
        float4 b = bias4[idx & bm4];
        v.x += b.x; v.y += b.y; v.z += b.z; v.w += b.w;
        if (doRelu) {
            v.x = fmaxf(v.x, 0.0f); v.y = fmaxf(v.y, 0.0f);
            v.z = fmaxf(v.z, 0.0f); v.w = fmaxf(v.w, 0.0f);
        }
        x4[idx] = v;
    }
}

// out = agg + b3 ; softmax over 64 cols iff mode != 1. One wave per row (wave32: 2 cols/lane).
__global__ void k_out_finalize(const float* __restrict__ agg, const float* __restrict__ b3,
                               const int* __restrict__ mode, float* __restrict__ out, int n) {
    int wid  = (blockIdx.x * blockDim.x + threadIdx.x) >> 5;
    int lane = threadIdx.x & 31;
    if (wid >= n) return;
    const float* row = agg + (size_t)wid * 64;
    float v0 = row[lane]      + b3[lane];
    float v1 = row[lane + 32] + b3[lane + 32];
    if (mode[0] != 1) {
        float m = fmaxf(v0, v1);
        for (int o = 16; o > 0; o >>= 1) m = fmaxf(m, __shfl_xor(m, o));
        float e0 = __expf(v0 - m), e1 = __expf(v1 - m);
        float s = e0 + e1;
        for (int o = 16; o > 0; o >>= 1) s += __shfl_xor(s, o);
        float inv = 1.0f / s;
        v0 = e0 * inv; v1 = e1 * inv;
    }
    out[(size_t)wid * 64 + lane]      = v0;
    out[(size_t)wid * 64 + lane + 32] = v1;
}

// ---------------- host launch ----------------

static inline int cdiv(long a, long b) { return (int)((a + b - 1) / b); }

extern "C" void kernel_launch(void* const* d_in, const int* in_sizes, int n_in,
                              void* d_out, int out_size, void* d_ws, size_t ws_size,
                              hipStream_t stream) {
    const int   N = NNODES, E = NEDGES;
    const float* x    = (const float*)d_in[0];
    const float* W1   = (const float*)d_in[1];
    const float* b1   = (const float*)d_in[2];
    const float* W2   = (const float*)d_in[3];
    const float* b2   = (const float*)d_in[4];
    const float* L1w  = (const float*)d_in[5];
    const float* L1b  = (const float*)d_in[6];
    const float* L2w  = (const float*)d_in[7];
    const float* L2b  = (const float*)d_in[8];
    const float* W3   = (const float*)d_in[9];
    const float* b3   = (const float*)d_in[10];
    const int*   adj  = (const int*)d_in[11];   // [2,E]: row0=src, row1=dst
    const int*   adj2 = (const int*)d_in[12];
    const int*   mode = (const int*)d_in[13];

    float* bufA = (float*)d_ws;                       // N*256 f32
    float* bufB = bufA + (size_t)N * 256;             // N*256 f32
    float* dinv = bufB + (size_t)N * 256;             // N f32

    const int TPB = 256;
    const dim3 blk(TPB);
    const int gemmGX = cdiv((long)N / 16, 8);         // 3125 m-tiles / 8 waves = 391
    const int v4Blocks256 = cdiv((long)N * 64, TPB);  // N*256/4 float4 elems
    const int v4Blocks64  = cdiv((long)N * 16, TPB);  // N*64/4
    const int edgeWaveBlk = cdiv((long)E * 32, TPB);

    // ---- conv1: adj, x(128) @ W1 -> 256, relu ----
    k_init_deg <<<cdiv(N, TPB), blk, 0, stream>>>(dinv, N);
    k_count_deg<<<cdiv(E, TPB), blk, 0, stream>>>(dinv, adj + E, E);
    k_rsqrt_deg<<<cdiv(N, TPB), blk, 0, stream>>>(dinv, N);
    k_gemm_wmma<<<dim3(gemmGX, 4), blk, 128 * 64 * 4, stream>>>(x, W1, nullptr, bufA, N, 128, 256, 0);
    k_agg_init <<<v4Blocks256, blk, 0, stream>>>((const float4*)bufA, dinv, (float4*)bufB, N * 64, 6);
    k_agg_edges<<<edgeWaveBlk, blk, 0, stream>>>(bufA, dinv, adj, adj + E, bufB, E, 256);
    k_bias_relu<<<v4Blocks256, blk, 0, stream>>>((float4*)bufB, (const float4*)b1, N * 64, 63, 1);

    // ---- conv2: adj2, h(256) @ W2 -> 256, relu ----
    k_init_deg <<<cdiv(N, TPB), blk, 0, stream>>>(dinv, N);
    k_count_deg<<<cdiv(E, TPB), blk, 0, stream>>>(dinv, adj2 + E, E);
    k_rsqrt_deg<<<cdiv(N, TPB), blk, 0, stream>>>(dinv, N);
    k_gemm_wmma<<<dim3(gemmGX, 4), blk, 256 * 64 * 4, stream>>>(bufB, W2, nullptr, bufA, N, 256, 256, 0);
    k_agg_init <<<v4Blocks256, blk, 0, stream>>>((const float4*)bufA, dinv, (float4*)bufB, N * 64, 6);
    k_agg_edges<<<edgeWaveBlk, blk, 0, stream>>>(bufA, dinv, adj2, adj2 + E, bufB, E, 256);
    k_bias_relu<<<v4Blocks256, blk, 0, stream>>>((float4*)bufB, (const float4*)b2, N * 64, 63, 1);

    // ---- lin1: relu(h @ L1w + L1b)  256 -> 128 ----
    k_gemm_wmma<<<dim3(gemmGX, 2), blk, 256 * 64 * 4, stream>>>(bufB, L1w, L1b, bufA, N, 256, 128, 1);
    // ---- lin2: relu(h @ L2w + L2b)  128 -> 256 ----
    k_gemm_wmma<<<dim3(gemmGX, 4), blk, 128 * 64 * 4, stream>>>(bufA, L2w, L2b, bufB, N, 128, 256, 1);

    // ---- conv3: adj2 (dinv still valid), h(256) @ W3 -> 64, + b3, optional softmax ----
    k_gemm_wmma<<<dim3(gemmGX, 1), blk, 256 * 64 * 4, stream>>>(bufB, W3, nullptr, bufA, N, 256, 64, 0);
    k_agg_init <<<v4Blocks64, blk, 0, stream>>>((const float4*)bufA, dinv, (float4*)bufB, N * 16, 4);
    k_agg_edges<<<edgeWaveBlk, blk, 0, stream>>>(bufA, dinv, adj2, adj2 + E, bufB, E, 64);
    k_out_finalize<<<cdiv((long)N * 32, TPB), blk, 0, stream>>>(bufB, b3, mode, (float*)d_out, N);
}